// SNRTransGate_34402688041279
// MI455X (gfx1250) — compile-verified
//
#include <hip/hip_runtime.h>

// SNRTransGate on MI455X (gfx1250): gate z[o,n] folded into B, one fp32 GEMM
// C[8192x2048] = A[8192x2048] * W[2048x2048] via V_WMMA_F32_16X16X4_F32.
// HBM-bound (~150MB -> ~6.5us @ 23.3TB/s); fp32 WMMA >> required ~10.7 TF/s.
// A tile: async DMA to LDS (global_load_async_to_lds_b128, double-buffered).
// B tile: register-prefetched, z-scaled, stored in K-paired LDS layout so each
// WMMA B fragment is a single aligned ds_load_b64 (no repack movs).

typedef __attribute__((ext_vector_type(2))) float v2f;
typedef __attribute__((ext_vector_type(4))) float v4f;
typedef __attribute__((ext_vector_type(8))) float v8f;
typedef int b128i __attribute__((vector_size(16)));   // builtin's b128 unit type

#define AS1 __attribute__((address_space(1)))
#define AS3 __attribute__((address_space(3)))

#if defined(__has_builtin)
#  if __has_builtin(__builtin_amdgcn_global_load_async_to_lds_b128)
#    define HAVE_ASYNC_LDS 1
#  else
#    define HAVE_ASYNC_LDS 0
#  endif
#else
#  define HAVE_ASYNC_LDS 0
#endif

#define NIN     8
#define NOUT    8
#define UNITS   256
#define BATCH   8192
#define K_TOTAL 2048
#define COLS    2048

#define TM 128
#define TN 128
#define TK 32
#define NCHUNK (K_TOTAL / TK)   // 64

#define A_PITCH  36    // dwords/row: 144B (16B aligned), conflict-free frag reads
#define B_PPITCH 288   // dwords per K-pair row: 256 data + 32 pad (bank shift 32)

__device__ __forceinline__ void wait_async0() {
#if HAVE_ASYNC_LDS
#  if __has_builtin(__builtin_amdgcn_s_wait_asynccnt)
    __builtin_amdgcn_s_wait_asynccnt(0);
#  else
    asm volatile("s_wait_asynccnt 0x0" ::: "memory");
#  endif
#endif
}

// A tile (raw, 128x32 f32) -> LDS buffer. Async DMA when available.
__device__ __forceinline__ void stage_A(const float* __restrict__ x,
                                        int row_base, int k0,
                                        float* lds, int wave, int lane) {
    const int mrow = wave * 16 + (lane >> 3);
    const int kk4  = (lane & 7) << 2;
#pragma unroll
    for (int i = 0; i < 4; ++i) {
        const int m = mrow + i * 4;
        const float* gp = x + (size_t)(row_base + m) * K_TOTAL + k0 + kk4;
        float* lp = lds + m * A_PITCH + kk4;
#if HAVE_ASYNC_LDS
        __builtin_amdgcn_global_load_async_to_lds_b128(
            (AS1 b128i*)(unsigned long long)(const void*)gp,
            (AS3 b128i*)(unsigned int)(unsigned long long)(void*)lp,
            /*offset=*/0, /*cpol=*/0);
#else
        *(v4f*)lp = *(const v4f*)gp;
#endif
    }
}

struct Bregs { v4f r0[2]; v4f r1[2]; };

// Prefetch B tile (32x128 of trans[o][n]) into registers (2 row-pairs/thread).
__device__ __forceinline__ void prefetch_B(const float* __restrict__ trans,
                                           size_t tbase, int u0, int vbase,
                                           int tid, Bregs& br) {
#pragma unroll
    for (int i = 0; i < 2; ++i) {
        const int idx = tid + i * 256;
        const int kp  = idx >> 5;              // K pair 0..15
        const int n4  = (idx & 31) << 2;       // N 0..124
        const float* gp = trans + tbase + (size_t)(u0 + 2 * kp) * UNITS + vbase + n4;
        br.r0[i] = *(const v4f*)gp;            // row 2kp
        br.r1[i] = *(const v4f*)(gp + UNITS);  // row 2kp+1
    }
}

// Scale by gate z and store to LDS in K-paired layout: (k,N) -> kp*PP + 2N + (k&1)
__device__ __forceinline__ void commit_B(float* bsm, int tid, float zz,
                                         const Bregs& br) {
#pragma unroll
    for (int i = 0; i < 2; ++i) {
        const int idx = tid + i * 256;
        const int kp  = idx >> 5;
        const int n4  = (idx & 31) << 2;
        float* bp = bsm + kp * B_PPITCH + n4 * 2;
#pragma unroll
        for (int c = 0; c < 4; ++c) {
            v2f pr;
            pr[0] = br.r0[i][c] * zz;
            pr[1] = br.r1[i][c] * zz;
            *(v2f*)(bp + 2 * c) = pr;          // aligned b64 store
        }
    }
}

__global__ __launch_bounds__(256) void snr_gemm_kernel(
    const float* __restrict__ x,        // [8192, 8, 256] == A[8192,2048]
    const float* __restrict__ alpha,    // [1]
    const float* __restrict__ u_param,  // [8(o), 8(n)]
    const float* __restrict__ trans,    // [8(o), 8(n), 256(u), 256(v)]
    float* __restrict__ out)            // [8192, 8, 256] == C[8192,2048]
{
    __shared__ float Asm[2][TM * A_PITCH];          // 2 * 18,432B
    __shared__ float Bsm[(TK / 2) * B_PPITCH];      // 18,432B  (total 55,296B)

    const int tid  = threadIdx.x;
    const int lane = tid & 31;
    const int wave = tid >> 5;
    const int wrow = wave & 3;     // 4 wave-rows * 32 M
    const int wcol = wave >> 2;    // 2 wave-cols * 64 N
    const int half = lane >> 4;
    const int r    = lane & 15;

    const int col_base = blockIdx.x * TN;
    const int row_base = blockIdx.y * TM;
    const int o     = col_base / UNITS;
    const int vbase = col_base - o * UNITS;

    const float log_alpha_beta = __logf(alpha[0]) / 0.9f;   // log(alpha)/BETA
    const float* up = u_param + o * NIN;

    v8f acc[2][4];
#pragma unroll
    for (int i = 0; i < 2; ++i)
#pragma unroll
        for (int j = 0; j < 4; ++j)
#pragma unroll
            for (int e = 0; e < 8; ++e)
                acc[i][j][e] = 0.0f;

    // ---- prologue: chunk 0 ----
    {
        const float u = up[0];
        const float s = 1.0f / (1.0f + __expf(-(__logf(u) - __logf(1.0f - u)
                                                + log_alpha_beta)));
        const float z = fminf(fmaxf(s * 1.2f - 0.1f, 0.0f), 1.0f);
        stage_A(x, row_base, 0, &Asm[0][0], wave, lane);
        Bregs br;
        prefetch_B(trans, (size_t)o * NIN * UNITS * UNITS, 0, vbase, tid, br);
        commit_B(&Bsm[0], tid, z, br);
        wait_async0();
        __syncthreads();
    }

    for (int kc = 0; kc < NCHUNK; ++kc) {
        const int  cur  = kc & 1;
        const int  nxt  = cur ^ 1;
        const bool more = (kc + 1) < NCHUNK;

        Bregs br;
        float z1 = 0.0f;
        if (more) {
            const int k1 = (kc + 1) * TK;
            const int n1 = k1 >> 8;
            const float u = up[n1];
            const float s = 1.0f / (1.0f + __expf(-(__logf(u) - __logf(1.0f - u)
                                                    + log_alpha_beta)));
            z1 = fminf(fmaxf(s * 1.2f - 0.1f, 0.0f), 1.0f);
            stage_A(x, row_base, k1, &Asm[nxt][0], wave, lane);   // async DMA
            prefetch_B(trans, ((size_t)o * NIN + n1) * (UNITS * UNITS),
                       k1 & 255, vbase, tid, br);                 // global->regs
        }

        // ---- compute chunk kc: 8 K-steps x (2x4) WMMA tiles per wave ----
        const float* As = &Asm[cur][0];
#pragma unroll
        for (int ks = 0; ks < TK / 4; ++ks) {
            const int kk = ks * 4 + 2 * half;    // A fragment K base (even)
            const int kp = ks * 2 + half;        // B K-pair row
            v2f a[2], b[4];
#pragma unroll
            for (int mt = 0; mt < 2; ++mt)
                a[mt] = *(const v2f*)(As + (wrow * 32 + mt * 16 + r) * A_PITCH + kk);
#pragma unroll
            for (int nt = 0; nt < 4; ++nt)
                b[nt] = *(const v2f*)(&Bsm[0] + kp * B_PPITCH
                                      + (wcol * 64 + nt * 16 + r) * 2);
#pragma unroll
            for (int mt = 0; mt < 2; ++mt)
#pragma unroll
                for (int nt = 0; nt < 4; ++nt)
                    acc[mt][nt] = __builtin_amdgcn_wmma_f32_16x16x4_f32(
                        false, a[mt], false, b[nt],
                        (short)0, acc[mt][nt], false, false);
        }

        __syncthreads();                 // all waves done reading Bsm / Asm[cur]
        if (more) {
            commit_B(&Bsm[0], tid, z1, br);   // scaled B for chunk kc+1
            wait_async0();                    // Asm[nxt] DMA complete
        }
        __syncthreads();
    }

    // ---- write back: C/D layout VGPR j -> M = 8*half + j, N = r ----
#pragma unroll
    for (int mt = 0; mt < 2; ++mt) {
#pragma unroll
        for (int nt = 0; nt < 4; ++nt) {
            const int ccol = col_base + wcol * 64 + nt * 16 + r;
#pragma unroll
            for (int j = 0; j < 8; ++j) {
                const int crow = row_base + wrow * 32 + mt * 16 + half * 8 + j;
                out[(size_t)crow * COLS + ccol] = acc[mt][nt][j];
            }
        }
    }
}

extern "C" void kernel_launch(void* const* d_in, const int* in_sizes, int n_in,
                              void* d_out, int out_size, void* d_ws, size_t ws_size,
                              hipStream_t stream) {
    (void)in_sizes; (void)n_in; (void)out_size; (void)d_ws; (void)ws_size;
    const float* x       = (const float*)d_in[0];
    const float* alpha   = (const float*)d_in[1];
    const float* u_param = (const float*)d_in[2];
    const float* trans   = (const float*)d_in[3];
    float* out           = (float*)d_out;

    dim3 grid(COLS / TN, BATCH / TM);   // 16 x 64 = 1024 blocks
    snr_gemm_kernel<<<grid, 256, 0, stream>>>(x, alpha, u_param, trans, out);
}